// HashEncoding_74809740362341
// MI455X (gfx1250) — compile-verified
//
#include <hip/hip_runtime.h>
#include <stdint.h>

#define TABLE_BITS 19
#define TABLE_MASK ((1u << TABLE_BITS) - 1u)
#define P1 2654435761u
#define P2 805459861u
#define P3 3674653429u

#define NLDS_LEVELS 8
#define LDS_ENTRIES 10371   // sum of (floor(1.5^l)+2)^3 for l=0..7

// 1.5^l, exact in f32 for l<=15 (matches reference f64->f32 cast bit-exactly)
__device__ __constant__ float kScales[16] = {
    1.0f, 1.5f, 2.25f, 3.375f, 5.0625f, 7.59375f, 11.390625f, 17.0859375f,
    25.62890625f, 38.443359375f, 57.6650390625f, 86.49755859375f,
    129.746337890625f, 194.6195068359375f, 291.92926025390625f,
    437.89389038085938f};

// dense grid dim per low level: floor(1.5^l)+2  (per-lane dynamic use)
__device__ __constant__ int kDims[NLDS_LEVELS]     = {3, 3, 4, 5, 7, 9, 13, 19};
__device__ __constant__ int kOffs[NLDS_LEVELS + 1] = {0, 27, 54, 118, 243, 586, 1315, 3512, 10371};

typedef float v2f __attribute__((ext_vector_type(2)));

__global__ __launch_bounds__(256) void HashEncoding_kernel(
    const float* __restrict__ x, const float* __restrict__ tables,
    float* __restrict__ out, int npoints)
{
    __shared__ float2 smem[LDS_ENTRIES];  // 82,968 B: levels 0..7, densely indexed by corner coords

    const int tid = threadIdx.x;

    // ---------------- Preload levels 0..7 into LDS via the async gather engine ----------------
    // Unrolled per level so D is compile-time: divisions become mul/shift, no search loop.
    // Each lane hashes one dense corner and issues GLOBAL_LOAD_ASYNC_TO_LDS_B64 (ASYNCcnt).
    {
        const uint32_t ldsBase = (uint32_t)(uintptr_t)(&smem[0]);  // low 32 bits = LDS byte offset
        const uint64_t tabBase = (uint64_t)(uintptr_t)tables;
        const int DIMS[NLDS_LEVELS] = {3, 3, 4, 5, 7, 9, 13, 19};
        const int OFFS[NLDS_LEVELS] = {0, 27, 54, 118, 243, 586, 1315, 3512};
        #pragma unroll
        for (int l = 0; l < NLDS_LEVELS; ++l) {
            const int D = DIMS[l], D2 = D * D, D3 = D2 * D;
            const int off = OFFS[l];
            const uint32_t hl = (uint32_t)l * P3;
            for (int e = tid; e < D3; e += 256) {
                const int cz = e / D2;
                const int r  = e - cz * D2;
                const int cy = r / D;
                const int cx = r - cy * D;
                const uint32_t idx = ((uint32_t)cx ^ ((uint32_t)cy * P1)
                                      ^ ((uint32_t)cz * P2) ^ hl) & TABLE_MASK;
                const uint64_t gaddr = tabBase + ((((uint64_t)l << TABLE_BITS) + idx) << 3);
                const uint32_t laddr = ldsBase + ((uint32_t)(off + e) << 3);
                asm volatile("global_load_async_to_lds_b64 %0, %1, off"
                             :: "v"(laddr), "v"(gaddr) : "memory");
            }
        }
        asm volatile("s_wait_asynccnt 0" ::: "memory");
        __syncthreads();
    }

    // ---------------- Main persistent loop: one thread per (point, level-pair) ----------------
    // Thread handles levels (lA, lA+8): lA always LDS path, lA+8 always global path.
    // -> no EXEC-masked divergence; all 32 lanes active in both gather sections.
    const int lA = tid & 7;            // grid stride is a multiple of 8 -> invariant
    const int lB = lA + 8;

    const float scaleA = kScales[lA];
    const float scaleB = kScales[lB];
    const uint32_t hlB = (uint32_t)lB * P3;
    const float2* __restrict__ tlB = (const float2*)tables + ((size_t)lB << TABLE_BITS);
    const int D = kDims[lA], D2 = D * D;
    const int offA = kOffs[lA];

    const long long total  = (long long)npoints << 3;
    const long long stride = (long long)gridDim.x * (long long)blockDim.x;

    for (long long q = (long long)blockIdx.x * blockDim.x + tid; q < total; q += stride) {
        const int point = (int)(q >> 3);
        const float* xp = x + (size_t)point * 3;

        const float ux = fminf(fmaxf(xp[0], 0.0f), 1.0f);
        const float uy = fminf(fmaxf(xp[1], 0.0f), 1.0f);
        const float uz = fminf(fmaxf(xp[2], 0.0f), 1.0f);

        // ---- low level lA: dense LDS lookup ----
        float rA0 = 0.0f, rA1 = 0.0f;
        {
            const float px = ux * scaleA, py = uy * scaleA, pz = uz * scaleA;
            const float gx = floorf(px), gy = floorf(py), gz = floorf(pz);
            const float fx = px - gx, fy = py - gy, fz = pz - gz;
            const int base = offA + (int)gx + (int)gy * D + (int)gz * D2;
            const float wx1 = fx, wx0 = 1.0f - fx;
            const float wy1 = fy, wy0 = 1.0f - fy;
            const float wz1 = fz, wz0 = 1.0f - fz;
            #pragma unroll
            for (int c = 0; c < 8; ++c) {
                const int li = base + (c & 1) + ((c >> 1) & 1) * D + ((c >> 2) & 1) * D2;
                const float2 f = smem[li];            // pairs merge into ds_load_b128
                const float w = ((c & 1) ? wx1 : wx0) * ((c & 2) ? wy1 : wy0)
                              * ((c & 4) ? wz1 : wz0);
                rA0 = fmaf(f.x, w, rA0);
                rA1 = fmaf(f.y, w, rA1);
            }
        }

        // ---- high level lB: uint32 hash + L2-resident global gather ----
        float rB0 = 0.0f, rB1 = 0.0f;
        {
            const float px = ux * scaleB, py = uy * scaleB, pz = uz * scaleB;
            const float gx = floorf(px), gy = floorf(py), gz = floorf(pz);
            const float fx = px - gx, fy = py - gy, fz = pz - gz;
            const uint32_t hx0 = (uint32_t)(int)gx,      hx1 = hx0 + 1u;
            const uint32_t hy0 = (uint32_t)(int)gy * P1, hy1 = hy0 + P1;
            const uint32_t hz0 = (uint32_t)(int)gz * P2, hz1 = hz0 + P2;
            const float wx1 = fx, wx0 = 1.0f - fx;
            const float wy1 = fy, wy0 = 1.0f - fy;
            const float wz1 = fz, wz0 = 1.0f - fz;
            #pragma unroll
            for (int c = 0; c < 8; ++c) {
                const uint32_t h = ((c & 1) ? hx1 : hx0) ^ ((c & 2) ? hy1 : hy0)
                                 ^ ((c & 4) ? hz1 : hz0) ^ hlB;
                const float2 f = tlB[h & TABLE_MASK];
                const float w = ((c & 1) ? wx1 : wx0) * ((c & 2) ? wy1 : wy0)
                              * ((c & 4) ? wz1 : wz0);
                rB0 = fmaf(f.x, w, rB0);
                rB1 = fmaf(f.y, w, rB1);
            }
        }

        // NT stores: 256 MB output stream must not evict the 64 MB L2-resident tables.
        float* orow = out + ((size_t)point << 5);
        v2f vA; vA.x = rA0; vA.y = rA1;
        v2f vB; vB.x = rB0; vB.y = rB1;
        __builtin_nontemporal_store(vA, (v2f*)(orow + (lA << 1)));
        __builtin_nontemporal_store(vB, (v2f*)(orow + (lB << 1)));
    }
}

extern "C" void kernel_launch(void* const* d_in, const int* in_sizes, int n_in,
                              void* d_out, int out_size, void* d_ws, size_t ws_size,
                              hipStream_t stream) {
    const float* x      = (const float*)d_in[0];   // [N,3] f32
    const float* tables = (const float*)d_in[1];   // [16, 2^19, 2] f32
    float* out          = (float*)d_out;           // [N,32] f32
    const int npoints   = in_sizes[0] / 3;

    // Persistent grid: 83 KB LDS preload amortized over ~32 level-pairs/thread.
    dim3 grid(2048), block(256);
    hipLaunchKernelGGL(HashEncoding_kernel, grid, block, 0, stream,
                       x, tables, out, npoints);
}